// GCN_56238301774262
// MI455X (gfx1250) — compile-verified
//
#include <hip/hip_runtime.h>

// ---------------------------------------------------------------------------
// GCN (3-layer, eval-mode BN) for MI455X / gfx1250, wave32.
// FP32 WMMA (V_WMMA_F32_16X16X4_F32) for the dense GEMMs (exact fp32 math;
// workload is aggregation/memory bound so low-precision WMMA buys nothing).
// ---------------------------------------------------------------------------

typedef float v2f __attribute__((ext_vector_type(2)));
typedef float v8f __attribute__((ext_vector_type(8)));

#define K_DIM 128   // inner dimension for every GEMM in this net

// ---------------------------------------------------------------------------
// Degree / dinv kernels
// ---------------------------------------------------------------------------
__global__ void k_fill(float* __restrict__ p, float v, int n) {
    int i = blockIdx.x * blockDim.x + threadIdx.x;
    if (i < n) p[i] = v;
}

__global__ void k_count_deg(const int* __restrict__ dst, float* __restrict__ deg, int E) {
    int e = blockIdx.x * blockDim.x + threadIdx.x;
    if (e < E) atomicAdd(&deg[dst[e]], 1.0f);
}

__global__ void k_rsqrt_inplace(float* __restrict__ p, int n) {
    int i = blockIdx.x * blockDim.x + threadIdx.x;
    if (i < n) p[i] = rsqrtf(p[i]);
}

// ---------------------------------------------------------------------------
// WMMA GEMM: C[M x NC] = A[M x 128] * W[128 x NC], fp32, row-major.
// One block per 16-row tile. blockDim = 32 * ceil(NC/16). Wave w computes the
// 16x16 output tile at columns [16w, 16w+16). A tile staged in padded LDS.
// ---------------------------------------------------------------------------
template <int NC>
__global__ void gemm_wmma_f32(const float* __restrict__ A,
                              const float* __restrict__ W,
                              float* __restrict__ C, int n_rows) {
    constexpr int NT  = (NC + 15) / 16;   // waves / n-tiles per block
    constexpr int LDA = K_DIM + 4;        // LDS row stride (bank-conflict-free)
    __shared__ float As[16 * LDA];

    const int row0 = blockIdx.x * 16;
    const int tid  = threadIdx.x;
    const int nthr = 32 * NT;

    // Cooperative, coalesced load of the 16 x 128 A tile into LDS (float4).
    for (int i = tid; i < 16 * (K_DIM / 4); i += nthr) {
        int r  = i / (K_DIM / 4);
        int c4 = i % (K_DIM / 4);
        float4 v = make_float4(0.f, 0.f, 0.f, 0.f);
        if (row0 + r < n_rows)
            v = ((const float4*)(A + (size_t)(row0 + r) * K_DIM))[c4];
        *(float4*)&As[r * LDA + c4 * 4] = v;
    }
    __syncthreads();

    const int wave = tid >> 5;          // n-tile index
    const int lane = tid & 31;
    const int m    = lane & 15;         // A row within tile
    const int kh   = (lane >> 4) * 2;   // lanes 0-15: K={0,1}; lanes 16-31: K={2,3}
    const int col  = wave * 16 + (lane & 15);

    v8f acc = {};
    #pragma unroll
    for (int k = 0; k < K_DIM; k += 4) {
        v2f a;
        // 8-byte LDS read; pad makes banks 4m+kh .. 4m+kh+1 all distinct.
        a.x = As[m * LDA + k + kh];
        a.y = As[m * LDA + k + kh + 1];
        v2f b;
        float bx = 0.f, by = 0.f;
        if ((NC % 16 == 0) || (col < NC)) {      // reconverges before WMMA
            bx = W[(size_t)(k + kh) * NC + col];
            by = W[(size_t)(k + kh + 1) * NC + col];
        }
        b.x = bx; b.y = by;
        acc = __builtin_amdgcn_wmma_f32_16x16x4_f32(
            /*neg_a=*/false, a, /*neg_b=*/false, b,
            /*c_mod=*/(short)0, acc, /*reuse_a=*/false, /*reuse_b=*/false);
    }

    // D layout: VGPR j -> row (lane<16 ? j : 8+j), col = lane&15 within tile.
    const int mbase = (lane >> 4) * 8;
    if ((NC % 16 == 0) || (col < NC)) {
        #pragma unroll
        for (int j = 0; j < 8; ++j) {
            int r = row0 + mbase + j;
            if (r < n_rows) C[(size_t)r * NC + col] = acc[j];
        }
    }
}

// ---------------------------------------------------------------------------
// agg[i,f] = h[i,f] * dinv[i]^2 + bias[f]   (self-loop term + bias, full init)
// ---------------------------------------------------------------------------
template <int F>
__global__ void k_self_init(const float* __restrict__ h,
                            const float* __restrict__ dinv,
                            const float* __restrict__ bias,
                            float* __restrict__ agg, int n_nodes) {
    int gid = blockIdx.x * blockDim.x + threadIdx.x;
    if (gid >= n_nodes * F) return;
    int i = gid / F;
    int f = gid % F;
    float di = dinv[i];
    agg[gid] = h[gid] * (di * di) + bias[f];
}

// ---------------------------------------------------------------------------
// Edge scatter: one wave32 per edge. agg[dst,f] += h[src,f] * dinv[s]*dinv[d]
// ---------------------------------------------------------------------------
template <int F>
__global__ void k_scatter(const int* __restrict__ src, const int* __restrict__ dst,
                          const float* __restrict__ dinv,
                          const float* __restrict__ h,
                          float* __restrict__ agg, int E) {
    int wid  = (blockIdx.x * blockDim.x + threadIdx.x) >> 5;  // wave32
    int lane = threadIdx.x & 31;
    if (wid >= E) return;                                     // wave-uniform exit
    int s = src[wid], d = dst[wid];
    float w = dinv[s] * dinv[d];
    const float* hs = h   + (size_t)s * F;
    float*       ad = agg + (size_t)d * F;
    if constexpr (F == 128) {
        int f0 = lane * 4;
        float4 v = *(const float4*)(hs + f0);
        atomicAdd(ad + f0 + 0, v.x * w);
        atomicAdd(ad + f0 + 1, v.y * w);
        atomicAdd(ad + f0 + 2, v.z * w);
        atomicAdd(ad + f0 + 3, v.w * w);
    } else {
        for (int f = lane; f < F; f += 32)
            atomicAdd(ad + f, hs[f] * w);
    }
}

// ---------------------------------------------------------------------------
// h = relu(h) * (rsqrt(1+eps)*gamma[f]) + beta[f]   (eval BatchNorm, in place)
// ---------------------------------------------------------------------------
__global__ void k_relu_bn(float* __restrict__ h,
                          const float* __restrict__ gamma,
                          const float* __restrict__ beta, int n_nodes) {
    int gid = blockIdx.x * blockDim.x + threadIdx.x;
    if (gid >= n_nodes * 128) return;
    int f = gid & 127;
    const float inv = rsqrtf(1.0f + 1e-5f);
    float v = h[gid];
    v = fmaxf(v, 0.0f);
    h[gid] = v * (inv * gamma[f]) + beta[f];
}

// ---------------------------------------------------------------------------
// Launcher
// ---------------------------------------------------------------------------
extern "C" void kernel_launch(void* const* d_in, const int* in_sizes, int n_in,
                              void* d_out, int out_size, void* d_ws, size_t ws_size,
                              hipStream_t stream) {
    const float* x      = (const float*)d_in[0];
    const int*   eidx   = (const int*)  d_in[1];
    const float* W1     = (const float*)d_in[2];
    const float* b1     = (const float*)d_in[3];
    const float* gamma1 = (const float*)d_in[4];
    const float* beta1  = (const float*)d_in[5];
    const float* W2     = (const float*)d_in[6];
    const float* b2     = (const float*)d_in[7];
    const float* gamma2 = (const float*)d_in[8];
    const float* beta2  = (const float*)d_in[9];
    const float* W3     = (const float*)d_in[10];
    const float* b3     = (const float*)d_in[11];
    float*       out    = (float*)d_out;

    const int N = in_sizes[0] / 128;   // 50000
    const int E = in_sizes[1] / 2;     // 800000
    const int* src = eidx;             // edge_index[0]
    const int* dst = eidx + E;         // edge_index[1]

    float* bufA = (float*)d_ws;                 // N*128 (GEMM output h)
    float* bufB = bufA + (size_t)N * 128;       // N*128 (aggregated / next input)
    float* dinv = bufB + (size_t)N * 128;       // N

    const int TPB = 256;
    const int gemm_blocks = (N + 15) / 16;
    const int elem128     = (N * 128 + TPB - 1) / TPB;
    const int elem40      = (N * 40  + TPB - 1) / TPB;
    const int scat_blocks = (E * 32 + TPB - 1) / TPB;   // one wave32 per edge

    // degrees -> dinv = rsqrt(in_deg + 1)
    k_fill<<<(N + TPB - 1) / TPB, TPB, 0, stream>>>(dinv, 1.0f, N);
    k_count_deg<<<(E + TPB - 1) / TPB, TPB, 0, stream>>>(dst, dinv, E);
    k_rsqrt_inplace<<<(N + TPB - 1) / TPB, TPB, 0, stream>>>(dinv, N);

    // ---- Layer 1: h = X @ W1 ; agg ; relu+bn ----
    gemm_wmma_f32<128><<<gemm_blocks, 256, 0, stream>>>(x, W1, bufA, N);
    k_self_init<128><<<elem128, TPB, 0, stream>>>(bufA, dinv, b1, bufB, N);
    k_scatter<128><<<scat_blocks, TPB, 0, stream>>>(src, dst, dinv, bufA, bufB, E);
    k_relu_bn<<<elem128, TPB, 0, stream>>>(bufB, gamma1, beta1, N);

    // ---- Layer 2 ----
    gemm_wmma_f32<128><<<gemm_blocks, 256, 0, stream>>>(bufB, W2, bufA, N);
    k_self_init<128><<<elem128, TPB, 0, stream>>>(bufA, dinv, b2, bufB, N);
    k_scatter<128><<<scat_blocks, TPB, 0, stream>>>(src, dst, dinv, bufA, bufB, E);
    k_relu_bn<<<elem128, TPB, 0, stream>>>(bufB, gamma2, beta2, N);

    // ---- Layer 3 (OUT=40): result straight into d_out ----
    gemm_wmma_f32<40><<<gemm_blocks, 96, 0, stream>>>(bufB, W3, bufA, N);
    k_self_init<40><<<elem40, TPB, 0, stream>>>(bufA, dinv, b3, out, N);
    k_scatter<40><<<scat_blocks, TPB, 0, stream>>>(src, dst, dinv, bufA, out, E);
}